// StructureLoss_2723009265971
// MI455X (gfx1250) — compile-verified
//
#include <hip/hip_runtime.h>
#include <hip/hip_bf16.h>
#include <math.h>

typedef float v2f __attribute__((ext_vector_type(2)));
typedef float v8f __attribute__((ext_vector_type(8)));

#define IMG_H 512
#define IMG_W 512
#define NIMG  16
#define TILE  64
#define HALO  15
#define MS_ROWS 94          // TILE + 2*HALO
#define MS_COLS 94
#define MSTR    95          // LDS stride for staged mask (bank-friendly)
#define HSTR    72          // LDS stride for horizontal sums (conflict-free WMMA B fetch)
#define KLEN    46          // valid K window per 16-row subtile (16 + 30)

// ---------------------------------------------------------------------------
// Kernel 1: one block = one 64x64 output tile of one image.
//  - stage mask tile + halo into LDS via async global->LDS copies
//  - horizontal 31-tap sums via running-sum (8 outputs/thread) into LDS
//  - vertical 31-tap sums via V_WMMA_F32_16X16X4_F32 with banded-ones A
//  - elementwise loss terms (1 exp + 1 log + 1 rcp per pixel)
//  - deterministic block reduction -> per-block slot in d_ws
// ---------------------------------------------------------------------------
__global__ __launch_bounds__(256) void structure_loss_main(
    const float* __restrict__ pred, const float* __restrict__ mask,
    float* __restrict__ blocksums)
{
    __shared__ float ms[MS_ROWS * MSTR];   // staged mask (with halo)
    __shared__ float hs[MS_ROWS * HSTR];   // horizontal box sums
    __shared__ float red[8 * 4];           // per-wave partials

    const int tid  = threadIdx.x;
    const int img  = blockIdx.x >> 6;      // 64 tiles per image
    const int tile = blockIdx.x & 63;
    const int y0   = (tile >> 3) * TILE;
    const int x0   = (tile & 7) * TILE;

    const float* mimg = mask + (size_t)img * IMG_H * IMG_W;
    const float* pimg = pred + ((size_t)img * 2 + 1) * IMG_H * IMG_W; // channel 1

    // ---- stage mask tile + halo (zero padding outside image) ----
    for (int i = tid; i < MS_ROWS * MS_COLS; i += 256) {
        int r = i / MS_COLS, c = i - r * MS_COLS;
        int gy = y0 - HALO + r, gx = x0 - HALO + c;
        if (gy >= 0 && gy < IMG_H && gx >= 0 && gx < IMG_W) {
            unsigned lds = (unsigned)(size_t)&ms[r * MSTR + c]; // low 32 bits = LDS offset
            const float* gp = mimg + gy * IMG_W + gx;
            asm volatile("global_load_async_to_lds_b32 %0, %1, off"
                         :: "v"(lds), "v"(gp) : "memory");
        } else {
            ms[r * MSTR + c] = 0.0f;
        }
    }
    asm volatile("s_wait_asynccnt 0" ::: "memory");
    __syncthreads();

    // ---- horizontal 31-tap sums: hs[r][c] = sum_{j=0..30} ms[r][c+j] ----
    // 8 outputs per thread via running sum: read 38 floats, 31+14 adds.
    for (int job = tid; job < MS_ROWS * 8; job += 256) {
        const int r  = job >> 3;
        const int c0 = (job & 7) << 3;            // 0,8,...,56
        const float* row = &ms[r * MSTR + c0];    // reads cols c0 .. c0+37 (<= 93)
        float s = 0.0f;
        #pragma unroll
        for (int j = 0; j < 31; ++j) s += row[j];
        hs[r * HSTR + c0] = s;
        #pragma unroll
        for (int j = 0; j < 7; ++j) {
            s += row[31 + j] - row[j];
            hs[r * HSTR + c0 + 1 + j] = s;
        }
    }
    __syncthreads();

    // ---- vertical 31-tap via WMMA (banded-ones A), then elementwise loss ----
    const int lane  = tid & 31;
    const int wave  = tid >> 5;
    const int mn    = lane & 15;            // M for A-lane, N for B/D-lane
    const int khalf = (lane >> 4) << 1;     // 0 (lanes 0-15) or 2 (lanes 16-31)
    const int mrow  = (lane >> 4) << 3;     // D: M = vgpr + 0 or +8

    float sw = 0.0f, swb = 0.0f, si = 0.0f, su = 0.0f;

    for (int st = wave; st < 16; st += 8) {      // uniform per wave: 2 subtiles
        const int sy = (st >> 2) << 4;
        const int sx = (st & 3) << 4;
        v8f acc = {0.f, 0.f, 0.f, 0.f, 0.f, 0.f, 0.f, 0.f};
        #pragma unroll
        for (int kb = 0; kb < 48; kb += 4) {
            const int k0 = kb + khalf;
            const int k1 = k0 + 1;
            v2f a, bv;
            // A[m][k] = 1 iff output row m's window covers hs row sy+k (k in [m, m+30])
            a.x = (k0 >= mn && k0 <= mn + 30) ? 1.0f : 0.0f;
            a.y = (k1 >= mn && k1 <= mn + 30) ? 1.0f : 0.0f;
            bv.x = (k0 < KLEN) ? hs[(sy + k0) * HSTR + sx + mn] : 0.0f;
            bv.y = (k1 < KLEN) ? hs[(sy + k1) * HSTR + sx + mn] : 0.0f;
            acc = __builtin_amdgcn_wmma_f32_16x16x4_f32(
                false, a, false, bv, (short)0, acc, false, false);
        }
        #pragma unroll
        for (int v = 0; v < 8; ++v) {
            const int ly = sy + mrow + v;       // local output row
            const int lx = sx + mn;             // local output col
            const float box = acc[v] * (1.0f / 961.0f);
            const float mv  = ms[(ly + HALO) * MSTR + (lx + HALO)];
            const float p   = pimg[(size_t)(y0 + ly) * IMG_W + (x0 + lx)];
            const float weit = 1.0f + 5.0f * fabsf(box - mv);
            // one exp feeds both softplus and sigmoid
            const float e    = __expf(-fabsf(p));            // v_exp_f32 path
            const float sp   = __logf(1.0f + e);             // log1p(exp(-|p|))
            const float inv  = __builtin_amdgcn_rcpf(1.0f + e);
            const float bce  = fmaxf(p, 0.0f) - p * mv + sp;
            const float ps   = (p >= 0.0f) ? inv : e * inv;  // sigmoid(p)
            sw  += weit;
            swb += weit * bce;
            si  += ps * mv * weit;
            su  += (ps + mv) * weit;
        }
    }

    // ---- wave reduce (wave32) then deterministic cross-wave fold ----
    for (int off = 16; off > 0; off >>= 1) {
        sw  += __shfl_down(sw,  off, 32);
        swb += __shfl_down(swb, off, 32);
        si  += __shfl_down(si,  off, 32);
        su  += __shfl_down(su,  off, 32);
    }
    if (lane == 0) {
        red[wave * 4 + 0] = sw;
        red[wave * 4 + 1] = swb;
        red[wave * 4 + 2] = si;
        red[wave * 4 + 3] = su;
    }
    __syncthreads();
    if (tid == 0) {
        float a0 = 0.f, a1 = 0.f, a2 = 0.f, a3 = 0.f;
        for (int w = 0; w < 8; ++w) {           // fixed order: deterministic
            a0 += red[w * 4 + 0];
            a1 += red[w * 4 + 1];
            a2 += red[w * 4 + 2];
            a3 += red[w * 4 + 3];
        }
        blocksums[(size_t)blockIdx.x * 4 + 0] = a0;
        blocksums[(size_t)blockIdx.x * 4 + 1] = a1;
        blocksums[(size_t)blockIdx.x * 4 + 2] = a2;
        blocksums[(size_t)blockIdx.x * 4 + 3] = a3;
    }
}

// ---------------------------------------------------------------------------
// Kernel 2: deterministic finalize -> scalar loss
// ---------------------------------------------------------------------------
__global__ __launch_bounds__(64) void structure_loss_finalize(
    const float* __restrict__ blocksums, float* __restrict__ out)
{
    __shared__ float part[NIMG * 4];
    __shared__ float loss[NIMG];
    const int t = threadIdx.x;

    if (t < NIMG * 4) {
        const int img = t >> 2, q = t & 3;
        float s = 0.0f;
        for (int blk = 0; blk < 64; ++blk)      // fixed order: deterministic
            s += blocksums[(size_t)((img << 6) + blk) * 4 + q];
        part[img * 4 + q] = s;
    }
    __syncthreads();
    if (t < NIMG) {
        const float sw  = part[t * 4 + 0];
        const float swb = part[t * 4 + 1];
        const float si  = part[t * 4 + 2];
        const float su  = part[t * 4 + 3];
        const float wbce = swb / sw;
        const float wiou = 1.0f - (si + 1.0f) / (su - si + 1.0f);
        loss[t] = wbce + wiou;
    }
    __syncthreads();
    if (t == 0) {
        float s = 0.0f;
        for (int i = 0; i < NIMG; ++i) s += loss[i];
        out[0] = s * (1.0f / (float)NIMG);
    }
}

extern "C" void kernel_launch(void* const* d_in, const int* in_sizes, int n_in,
                              void* d_out, int out_size, void* d_ws, size_t ws_size,
                              hipStream_t stream)
{
    (void)in_sizes; (void)n_in; (void)out_size; (void)ws_size;
    const float* pred = (const float*)d_in[0];   // [16,2,512,512] f32
    const float* mask = (const float*)d_in[1];   // [16,512,512]  f32
    float* blocksums  = (float*)d_ws;            // 1024 blocks * 4 f32 = 16 KB

    structure_loss_main<<<NIMG * 64, 256, 0, stream>>>(pred, mask, blocksums);
    structure_loss_finalize<<<1, 64, 0, stream>>>(blocksums, (float*)d_out);
}